// GW_28767690949418
// MI455X (gfx1250) — compile-verified
//
#include <hip/hip_runtime.h>
#include <hip/hip_bf16.h>

#define BB 16
#define CC 128
#define TT 8192
#define KK 8
#define RR 1.0f

typedef float v2f __attribute__((ext_vector_type(2)));
typedef float v8f __attribute__((ext_vector_type(8)));

// Constant-lane broadcast: lowers to v_readlane_b32.
__device__ __forceinline__ float readlane_f(float v, int lane) {
    const int i = __builtin_amdgcn_readlane(__builtin_bit_cast(int, v), lane);
    return __builtin_bit_cast(float, i);
}

// ---------------------------------------------------------------------------
// Kernel 1: tensor-core inclusive scan of w rows, fully on the WMMA pipe.
// One wave per (b,k) row. Each 256-element chunk is a 16x16 matrix A
// (element e = m*16 + n).
//   Pass 1: P = A @ U            (U upper-tri ones: row-inclusive prefix)
//   Pass 2: D = Lstrict @ S + P  (S[k][n] = rowsum s[k]: cross-row offsets)
// Each pass is four chained V_WMMA_F32_16X16X4_F32 (K=4 slabs). The row sums
// move lanes via 16 v_readlane; carry chains across the 32 chunks.
// ---------------------------------------------------------------------------
__global__ __launch_bounds__(32) void GW_scan_rows(const float* __restrict__ w,
                                                   float* __restrict__ wc) {
    const int row  = blockIdx.x;            // b*K + k
    const int lane = threadIdx.x;           // 0..31
    const int ln   = lane & 15;             // A: row m ; B: col n ; C/D: col n
    const int hi   = lane >> 4;             // lane half
    const int cb   = hi * 2;                // K-pair base within each K=4 slab

    const float* src = w  + (size_t)row * TT;
    float*       dst = wc + (size_t)row * TT;

    // Constant fragments: bu = U slabs (B operand, pass 1);
    //                     al = Lstrict slabs (A operand, pass 2).
    v2f bu[4], al[4];
#pragma unroll
    for (int i = 0; i < 4; ++i) {
        const int k0 = 4 * i + cb;
        bu[i].x = (k0     <= ln) ? 1.0f : 0.0f;   // U[k][n] = (k <= n)
        bu[i].y = (k0 + 1 <= ln) ? 1.0f : 0.0f;
        al[i].x = (k0     <  ln) ? 1.0f : 0.0f;   // Lstrict[m][k] = (k < m)
        al[i].y = (k0 + 1 <  ln) ? 1.0f : 0.0f;
    }

    float carry = 0.0f;
    for (int chunk = 0; chunk < TT / 256; ++chunk) {
        const float* cs = src + chunk * 256;
        // A fragments: lane holds A[ln][4i+cb], A[ln][4i+cb+1]  (8B aligned)
        v2f a0 = *(const v2f*)(cs + ln * 16 + 0  + cb);
        v2f a1 = *(const v2f*)(cs + ln * 16 + 4  + cb);
        v2f a2 = *(const v2f*)(cs + ln * 16 + 8  + cb);
        v2f a3 = *(const v2f*)(cs + ln * 16 + 12 + cb);

        // Pass 1: row-inclusive prefix P = A @ U.
        v8f p = {};
        p = __builtin_amdgcn_wmma_f32_16x16x4_f32(false, a0, false, bu[0], (short)0, p, false, false);
        p = __builtin_amdgcn_wmma_f32_16x16x4_f32(false, a1, false, bu[1], (short)0, p, false, false);
        p = __builtin_amdgcn_wmma_f32_16x16x4_f32(false, a2, false, bu[2], (short)0, p, false, false);
        p = __builtin_amdgcn_wmma_f32_16x16x4_f32(false, a3, false, bu[3], (short)0, p, false, false);

        // Row sums s[m] = P[m][15]; rows 0..7 lane 15, rows 8..15 lane 31.
        float s[16];
#pragma unroll
        for (int g = 0; g < 8; ++g) s[g]     = readlane_f(p[g], 15);
#pragma unroll
        for (int g = 0; g < 8; ++g) s[8 + g] = readlane_f(p[g], 31);

        // B fragments of S (s[k] replicated across n): per-half row select.
        v2f bs[4];
#pragma unroll
        for (int i = 0; i < 4; ++i) {
            bs[i].x = hi ? s[4 * i + 2] : s[4 * i + 0];
            bs[i].y = hi ? s[4 * i + 3] : s[4 * i + 1];
        }

        // Pass 2: D = Lstrict @ S + P  (full 256-element inclusive scan).
        v8f d = p;
        d = __builtin_amdgcn_wmma_f32_16x16x4_f32(false, al[0], false, bs[0], (short)0, d, false, false);
        d = __builtin_amdgcn_wmma_f32_16x16x4_f32(false, al[1], false, bs[1], (short)0, d, false, false);
        d = __builtin_amdgcn_wmma_f32_16x16x4_f32(false, al[2], false, bs[2], (short)0, d, false, false);
        d = __builtin_amdgcn_wmma_f32_16x16x4_f32(false, al[3], false, bs[3], (short)0, d, false, false);

        // D layout: VGPR g -> row m = g (lanes<16) / g+8 (lanes>=16), col n = ln.
#pragma unroll
        for (int g = 0; g < 8; ++g) {
            const int m = hi ? (g + 8) : g;
            dst[chunk * 256 + m * 16 + ln] = d[g] + carry;
        }
        // D[15][15] = chunk total (VGPR 7, lane 31).
        carry += readlane_f(d[7], 31);
    }
}

// ---------------------------------------------------------------------------
// Kernel 2: in-place linear-ramp correction + scaling + pad mask.
// One block per (b,k) row; w0/wL read by every thread before the barrier, so
// the in-place update is race-free.
// ---------------------------------------------------------------------------
__global__ __launch_bounds__(256) void GW_finish_wp(float* __restrict__ buf,
                                                    const int* __restrict__ lengths) {
    const int row = blockIdx.x;             // b*K + k
    const int b   = row / KK;
    const int len = lengths[b];
    float* p = buf + (size_t)row * TT;

    const float w0  = p[0];
    const float wL  = p[len - 1];
    const float lm1 = (float)(len - 1);
    __syncthreads();

    const float scale = RR / (float)KK;
    for (int t = threadIdx.x; t < TT; t += 256) {
        const float tf  = (float)t / lm1;
        float val = (p[t] - ((1.0f - tf) * w0 + tf * wL)) * scale;
        p[t] = (t >= len) ? 0.0f : val;
    }
}

// ---------------------------------------------------------------------------
// Kernel 3: compose the K warps. One block per batch element; f lives in LDS
// (32 KB of the 320 KB WGP pool). Each of the 7 steps gathers 4 taps from LDS,
// holds results in registers, barriers, writes back.
// ---------------------------------------------------------------------------
__global__ __launch_bounds__(256) void GW_compose(const float* __restrict__ wp,
                                                  float* __restrict__ fout) {
    __shared__ float fs[TT];
    const int b   = blockIdx.x;
    const int tid = threadIdx.x;
    const float* wpb = wp + (size_t)b * KK * TT;

    for (int t = tid; t < TT; t += 256) fs[t] = (float)t + wpb[t];
    __syncthreads();

    for (int i = 1; i < KK; ++i) {
        const float* wpi = wpb + (size_t)i * TT;
        float res[TT / 256];
#pragma unroll
        for (int j = 0; j < TT / 256; ++j) {
            const int   t  = tid + j * 256;
            const float g  = (float)t + wpi[t];
            const float fl = floorf(g);
            const int   i0 = (int)fl;
            const float s  = g - fl;
            const float s2 = s * s, s3 = s2 * s;
            const float wA = -0.5f * s3 +        s2 - 0.5f * s;
            const float wB =  1.5f * s3 - 2.5f * s2 + 1.0f;
            const float wC = -1.5f * s3 + 2.0f * s2 + 0.5f * s;
            const float wD =  0.5f * s3 - 0.5f * s2;
            const int j0 = min(max(i0 - 1, 0), TT - 1);
            const int j1 = min(max(i0    , 0), TT - 1);
            const int j2 = min(max(i0 + 1, 0), TT - 1);
            const int j3 = min(max(i0 + 2, 0), TT - 1);
            res[j] = wA * fs[j0] + wB * fs[j1] + wC * fs[j2] + wD * fs[j3];
        }
        __syncthreads();
#pragma unroll
        for (int j = 0; j < TT / 256; ++j) fs[tid + j * 256] = res[j];
        __syncthreads();
    }

    for (int t = tid; t < TT; t += 256) fout[(size_t)b * TT + t] = fs[t];
}

// ---------------------------------------------------------------------------
// Kernel 4: final cubic gather of x at f. Weights computed once per (b,t) and
// reused across all 128 channels; stores along t are fully coalesced; the
// 4-tap reads are near-coalesced (f ~ t + bounded drift) and L2-resident
// (x = 64 MiB << 192 MB L2). Prefetch one channel ahead (global_prefetch_b8)
// to hide first-touch gather latency.
// ---------------------------------------------------------------------------
__global__ __launch_bounds__(256) void GW_final(const float* __restrict__ x,
                                                const float* __restrict__ f,
                                                const int* __restrict__ lengths,
                                                float* __restrict__ y) {
    const int b   = blockIdx.y;
    const int t   = blockIdx.x * 256 + threadIdx.x;
    const int len = lengths[b];

    const float ft = f[(size_t)b * TT + t];
    const float fl = floorf(ft);
    const int   i0 = (int)fl;
    const float s  = ft - fl;
    const float s2 = s * s, s3 = s2 * s;
    const float wA = -0.5f * s3 +        s2 - 0.5f * s;
    const float wB =  1.5f * s3 - 2.5f * s2 + 1.0f;
    const float wC = -1.5f * s3 + 2.0f * s2 + 0.5f * s;
    const float wD =  0.5f * s3 - 0.5f * s2;
    const int j0 = min(max(i0 - 1, 0), TT - 1);
    const int j1 = min(max(i0    , 0), TT - 1);
    const int j2 = min(max(i0 + 1, 0), TT - 1);
    const int j3 = min(max(i0 + 2, 0), TT - 1);
    // x is zero-masked beyond len; fold the mask into the taps.
    const float mA = (j0 < len) ? wA : 0.0f;
    const float mB = (j1 < len) ? wB : 0.0f;
    const float mC = (j2 < len) ? wC : 0.0f;
    const float mD = (j3 < len) ? wD : 0.0f;
    const bool pad = (t >= len);

    const float* xb = x + (size_t)b * CC * TT;
    float*       yb = y + (size_t)b * CC * TT;

#pragma unroll 4
    for (int c = 0; c < CC; ++c) {
        const float* xc = xb + (size_t)c * TT;
        if (c + 1 < CC) __builtin_prefetch(xc + TT + j1, 0, 0);
        const float r = mA * xc[j0] + mB * xc[j1] + mC * xc[j2] + mD * xc[j3];
        yb[(size_t)c * TT + t] = pad ? 0.0f : r;
    }
}

// ---------------------------------------------------------------------------
extern "C" void kernel_launch(void* const* d_in, const int* in_sizes, int n_in,
                              void* d_out, int out_size, void* d_ws, size_t ws_size,
                              hipStream_t stream) {
    const float* x    = (const float*)d_in[0];
    const float* w    = (const float*)d_in[1];
    const int*   lens = (const int*)d_in[2];
    float*       y    = (float*)d_out;

    float* wc = (float*)d_ws;                       // B*K*T floats (4 MiB), becomes wp in place
    float* fb = wc + (size_t)BB * KK * TT;          // B*T floats (0.5 MiB)

    GW_scan_rows<<<BB * KK, 32, 0, stream>>>(w, wc);
    GW_finish_wp<<<BB * KK, 256, 0, stream>>>(wc, lens);
    GW_compose<<<BB, 256, 0, stream>>>(wc, fb);
    dim3 g4(TT / 256, BB);
    GW_final<<<g4, 256, 0, stream>>>(x, fb, lens, y);
}